// MonoAttention_6390911336615
// MI455X (gfx1250) — compile-verified
//
#include <hip/hip_runtime.h>
#include <hip/hip_bf16.h>
#include <math.h>

#define B_   256
#define T_   2048
#define K_   8
#define RNN_ 1024
#define ATT_ 128
#define MEM_ 512

typedef __attribute__((ext_vector_type(2))) float v2f;
typedef __attribute__((ext_vector_type(4))) float v4f;
typedef __attribute__((ext_vector_type(8))) float v8f;

// ---------------------------------------------------------------------------
// Kernel A: MLP -> location parameters.
// Grid: B/16 blocks, 256 threads (8 wave32). Each block handles 16 batch rows.
// x@W1 done with V_WMMA_F32_16X16X4_F32: wave w owns N-tile [16w,16w+16).
// ---------------------------------------------------------------------------
__global__ __launch_bounds__(256)
void mlp_kernel(const float* __restrict__ x,     // (B, RNN)
                const float* __restrict__ prev,  // (B, 1, K)
                const float* __restrict__ W1,    // (RNN, ATT)
                const float* __restrict__ b1,
                const float* __restrict__ g1,
                const float* __restrict__ be1,
                const float* __restrict__ W2,    // (ATT, 3K)
                const float* __restrict__ b2,
                const float* __restrict__ g2,
                const float* __restrict__ be2,
                float* __restrict__ wsm_g,       // (B,K) workspace
                float* __restrict__ loc_g,       // (B,K) workspace
                float* __restrict__ std_g,       // (B,K) workspace
                float* __restrict__ out_loc)     // (B,K) output slice
{
  __shared__ float As[16][256];        // A chunk: 16 rows x 256 K  (16 KB)
  __shared__ float hbuf[16][ATT_];     // 8 KB
  __shared__ float tbuf[16][3 * K_];   // 1.5 KB

  const int tid  = threadIdx.x;
  const int lane = tid & 31;
  const int wave = tid >> 5;              // 0..7
  const int r0   = blockIdx.x * 16;       // batch-row base
  const int n0   = wave * 16;             // N-tile base for this wave
  const int lm   = lane & 15;             // M index (A frag) / N index (B frag)
  const int kh   = (lane >> 4) * 2;       // K-half select per ISA layout

  v8f c = {};                              // 16x16 fp32 accumulator tile

  for (int kc = 0; kc < RNN_; kc += 256) {
    // Cooperative stage of A chunk into LDS: 16 x 64 float4 = 1024 vec4.
    for (int i = tid; i < 16 * 64; i += 256) {
      int row  = i >> 6;
      int col4 = i & 63;
      v4f v = *(const v4f*)(x + (size_t)(r0 + row) * RNN_ + kc + col4 * 4);
      *(v4f*)&As[row][col4 * 4] = v;
    }
    __syncthreads();

    #pragma unroll 4
    for (int k = 0; k < 256; k += 4) {
      v2f a, b;
      // A 16x4 fp32 frag: lanes 0-15 -> K = k,k+1 ; lanes 16-31 -> K = k+2,k+3
      a.x = As[lm][k + kh];
      a.y = As[lm][k + kh + 1];
      // B 4x16 fp32 frag: N striped across lanes, same K-half split
      size_t gk = (size_t)(kc + k + kh);
      b.x = W1[gk * ATT_ + n0 + lm];
      b.y = W1[(gk + 1) * ATT_ + n0 + lm];
      c = __builtin_amdgcn_wmma_f32_16x16x4_f32(
              /*neg_a=*/false, a, /*neg_b=*/false, b,
              /*c_mod=*/(short)0, c, /*reuse_a=*/false, /*reuse_b=*/false);
    }
    __syncthreads();
  }

  // Scatter C tile to hbuf (+bias): VGPR i -> M = i (lanes 0-15) / i+8 (16-31)
  {
    const int mh = (lane >> 4) * 8;
    #pragma unroll
    for (int i = 0; i < 8; ++i)
      hbuf[i + mh][n0 + lm] = c[i] + b1[n0 + lm];
  }
  __syncthreads();

  // LayerNorm(128) + ReLU. Wave w handles rows 2w, 2w+1; 4 cols per lane.
  for (int rr = wave * 2; rr < wave * 2 + 2; ++rr) {
    float v[4];
    float s = 0.f;
    #pragma unroll
    for (int j = 0; j < 4; ++j) { v[j] = hbuf[rr][lane + 32 * j]; s += v[j]; }
    for (int off = 16; off; off >>= 1) s += __shfl_xor(s, off, 32);
    float mean = s * (1.f / ATT_);
    float q = 0.f;
    #pragma unroll
    for (int j = 0; j < 4; ++j) { float d = v[j] - mean; q += d * d; }
    for (int off = 16; off; off >>= 1) q += __shfl_xor(q, off, 32);
    float rstd = rsqrtf(q * (1.f / ATT_) + 1e-5f);
    #pragma unroll
    for (int j = 0; j < 4; ++j) {
      int col = lane + 32 * j;
      float y = (v[j] - mean) * rstd * g1[col] + be1[col];
      hbuf[rr][col] = fmaxf(y, 0.f);
    }
  }
  __syncthreads();

  // h @ W2 + b2 : 16 rows x 24 cols, dot length 128 (tiny).
  for (int idx = tid; idx < 16 * 24; idx += 256) {
    int row = idx / 24, col = idx - row * 24;
    float acc = b2[col];
    #pragma unroll 8
    for (int i = 0; i < ATT_; ++i) acc += hbuf[row][i] * W2[i * (3 * K_) + col];
    tbuf[row][col] = acc;
  }
  __syncthreads();

  // Per-row LN(24) + split into (w, delta, scale) -> softmax / sigmoid / softplus
  if (tid < 16) {
    const int b = r0 + tid;
    float m = 0.f;
    for (int j = 0; j < 24; ++j) m += tbuf[tid][j];
    m *= (1.f / 24.f);
    float q = 0.f;
    for (int j = 0; j < 24; ++j) { float d = tbuf[tid][j] - m; q += d * d; }
    float rstd = rsqrtf(q * (1.f / 24.f) + 1e-5f);

    float w[K_], mx = -1e30f;
    #pragma unroll
    for (int k = 0; k < K_; ++k) {
      w[k] = (tbuf[tid][k] - m) * rstd * g2[k] + be2[k];
      mx = fmaxf(mx, w[k]);
    }
    float se = 0.f;
    #pragma unroll
    for (int k = 0; k < K_; ++k) { w[k] = expf(w[k] - mx); se += w[k]; }
    float inv = 1.f / se;

    #pragma unroll
    for (int k = 0; k < K_; ++k) {
      wsm_g[b * K_ + k] = w[k] * inv;
      float delta = (tbuf[tid][K_ + k] - m) * rstd * g2[K_ + k] + be2[K_ + k];
      float loc   = prev[b * K_ + k] + 1.f / (1.f + expf(-delta));
      loc_g[b * K_ + k]  = loc;
      out_loc[b * K_ + k] = loc;
      float sc = (tbuf[tid][2 * K_ + k] - m) * rstd * g2[2 * K_ + k]
                 + be2[2 * K_ + k] + 10.f;
      float sp = (sc > 20.f) ? sc : log1pf(expf(sc));
      std_g[b * K_ + k] = sp;
    }
  }
}

// ---------------------------------------------------------------------------
// Kernel B: erf-window alignment + context. One block per batch row.
// Bandwidth-bound: streams memory[b] (4 MB) once with float4 loads + prefetch.
// ---------------------------------------------------------------------------
__global__ __launch_bounds__(256)
void attend_kernel(const float* __restrict__ memory,        // (B, T, MEM)
                   const unsigned char* __restrict__ mask,  // (B, T) bool
                   const float* __restrict__ wsm_g,
                   const float* __restrict__ loc_g,
                   const float* __restrict__ std_g,
                   float* __restrict__ out_ctx,             // (B, MEM)
                   float* __restrict__ out_align)           // (B, T)
{
  __shared__ float al[T_];          // 8 KB alignment row
  __shared__ float wk[K_], lk[K_], ik[K_], hk[K_];
  __shared__ v4f red[128];          // 2 KB partial-ctx exchange

  const int b   = blockIdx.x;
  const int tid = threadIdx.x;

  if (tid < K_) {
    wk[tid] = wsm_g[b * K_ + tid];
    lk[tid] = loc_g[b * K_ + tid];
    float s = std_g[b * K_ + tid];
    ik[tid] = 1.f / s;
    hk[tid] = 0.5f / s;
  }
  __syncthreads();

  // align[t] = sum_k wsm_k * 0.5*(erf(d+h) - erf(d-h)),  d=(loc-t)/std, h=0.5/std
  for (int t = tid; t < T_; t += 256) {
    float pos = (float)t;
    float a = 0.f;
    #pragma unroll
    for (int k = 0; k < K_; ++k) {
      float d = (lk[k] - pos) * ik[k];
      float h = hk[k];
      float z = 0.5f * (erff(d + h) - erff(d - h));
      a = fmaf(wk[k], z, a);
    }
    if (mask[(size_t)b * T_ + t]) a = 0.f;
    al[t] = a;
    out_align[(size_t)b * T_ + t] = a;
  }
  __syncthreads();

  // ctx[b,d] = sum_t align[t] * memory[b,t,d].  dv covers 128 float4 columns,
  // th selects the t half; two halves reduced through LDS at the end.
  const int dv = tid & 127;
  const int th = tid >> 7;
  const float* mp  = memory + ((size_t)b * T_ + (size_t)th * (T_ / 2)) * MEM_ + dv * 4;
  const float* alp = &al[th * (T_ / 2)];

  v4f acc = {};
  #pragma unroll 4
  for (int t = 0; t < T_ / 2; ++t) {
    v4f mv = *(const v4f*)mp;
    __builtin_prefetch(mp + 16 * MEM_, 0, 1);   // global_prefetch_b8, 16 rows ahead
    float a = alp[t];
    acc.x = fmaf(a, mv.x, acc.x);
    acc.y = fmaf(a, mv.y, acc.y);
    acc.z = fmaf(a, mv.z, acc.z);
    acc.w = fmaf(a, mv.w, acc.w);
    mp += MEM_;
  }

  if (th == 1) red[dv] = acc;
  __syncthreads();
  if (th == 0) {
    v4f r = red[dv];
    acc.x += r.x; acc.y += r.y; acc.z += r.z; acc.w += r.w;
    *(v4f*)(out_ctx + (size_t)b * MEM_ + dv * 4) = acc;
  }
}

// ---------------------------------------------------------------------------
extern "C" void kernel_launch(void* const* d_in, const int* in_sizes, int n_in,
                              void* d_out, int out_size, void* d_ws, size_t ws_size,
                              hipStream_t stream) {
  const float* x    = (const float*)d_in[0];   // attention_hidden_state (B,RNN)
  const float* mem  = (const float*)d_in[1];   // memory (B,T,MEM)
  const float* prev = (const float*)d_in[2];   // previous_location (B,1,K)
  const unsigned char* mask = (const unsigned char*)d_in[3];  // (B,T) bool
  const float* W1   = (const float*)d_in[4];
  const float* b1   = (const float*)d_in[5];
  const float* g1   = (const float*)d_in[6];
  const float* be1  = (const float*)d_in[7];
  const float* W2   = (const float*)d_in[8];
  const float* b2   = (const float*)d_in[9];
  const float* g2   = (const float*)d_in[10];
  const float* be2  = (const float*)d_in[11];

  float* out       = (float*)d_out;
  float* out_ctx   = out;                       // (B, MEM)
  float* out_align = out + (size_t)B_ * MEM_;   // (B, T)
  float* out_loc   = out + (size_t)B_ * MEM_ + (size_t)B_ * T_;  // (B, K)

  float* ws    = (float*)d_ws;
  float* wsm_g = ws;                 // (B,K)
  float* loc_g = ws + B_ * K_;       // (B,K)
  float* std_g = ws + 2 * B_ * K_;   // (B,K)

  mlp_kernel<<<B_ / 16, 256, 0, stream>>>(x, prev, W1, b1, g1, be1,
                                          W2, b2, g2, be2,
                                          wsm_g, loc_g, std_g, out_loc);
  attend_kernel<<<B_, 256, 0, stream>>>(mem, mask, wsm_g, loc_g, std_g,
                                        out_ctx, out_align);
}